// BilinearSeqAttnAction1_11175504904499
// MI455X (gfx1250) — compile-verified
//
#include <hip/hip_runtime.h>
#include <math.h>

#define B_ 32
#define L_ 1024
#define X_ 1024
#define Y_ 1024
#define A_ 16

typedef float v2f __attribute__((ext_vector_type(2)));
typedef float v8f __attribute__((ext_vector_type(8)));

// ---------- block reduction helpers (256 threads = 8 wave32) ----------
__device__ inline float block_max(float v, float* red) {
  #pragma unroll
  for (int off = 16; off; off >>= 1) v = fmaxf(v, __shfl_xor(v, off, 32));
  const int lane = threadIdx.x & 31, wv = threadIdx.x >> 5;
  if (lane == 0) red[wv] = v;
  __syncthreads();
  float r = red[0];
  #pragma unroll
  for (int i = 1; i < 8; ++i) r = fmaxf(r, red[i]);
  __syncthreads();
  return r;
}

__device__ inline float block_sum(float v, float* red) {
  #pragma unroll
  for (int off = 16; off; off >>= 1) v += __shfl_xor(v, off, 32);
  const int lane = threadIdx.x & 31, wv = threadIdx.x >> 5;
  if (lane == 0) red[wv] = v;
  __syncthreads();
  float r = red[0];
  #pragma unroll
  for (int i = 1; i < 8; ++i) r += red[i];
  __syncthreads();
  return r;
}

// ---------- Kernel A: fused 48x1024x1024 fp32 GEMM via V_WMMA_F32_16X16X4_F32
// Z[m,x] = sum_y Arow[m,y] * weight[y,x]; Arow = [wa_h^T (16 rows); y (32 rows)]
__global__ __launch_bounds__(32) void gemm48_wmma(
    const float* __restrict__ wa_h, const float* __restrict__ yin,
    const float* __restrict__ weight, float* __restrict__ Z) {
  const int lane  = threadIdx.x;
  const int mtile = blockIdx.y;      // 0..2  (M = 48)
  const int ntile = blockIdx.x;      // 0..63 (N = 1024)
  const int ml    = lane & 15;
  const int half  = lane >> 4;       // lanes 0-15 -> K{0,1}, lanes 16-31 -> K{2,3}
  const int m     = mtile * 16 + ml;
  const int n     = ntile * 16 + ml;
  const float* __restrict__ arow =
      (m < A_) ? (wa_h + (size_t)m * Y_) : (yin + (size_t)(m - A_) * Y_);

  v8f acc = {};
  for (int k0 = 0; k0 < Y_; k0 += 4) {
    const int k = k0 + 2 * half;
    v2f a, b;
    a.x = arow[k];                           // A[m, k]   (16x4, row in lane)
    a.y = arow[k + 1];                       // A[m, k+1]
    b.x = weight[(size_t)k * X_ + n];        // B[k,   n] (4x16, row across lanes)
    b.y = weight[(size_t)(k + 1) * X_ + n];  // B[k+1, n]
    acc = __builtin_amdgcn_wmma_f32_16x16x4_f32(false, a, false, b,
                                                (short)0, acc, false, false);
  }
  // D layout: lanes 0-15 hold M=0..7 in v0..v7; lanes 16-31 hold M=8..15
  const int mbase = mtile * 16 + 8 * half;
  #pragma unroll
  for (int v = 0; v < 8; ++v)
    Z[(size_t)(mbase + v) * X_ + ntile * 16 + ml] = acc[v];
}

// ---------- Kernel B1: score2[a,:] = softmax(Z[a,:]) over X, one block per a
__global__ __launch_bounds__(256) void softmax_rows(const float* __restrict__ Z,
                                                    float* __restrict__ score2) {
  __shared__ float red[8];
  const int a = blockIdx.x;
  const int tid = threadIdx.x;
  const float* row = Z + (size_t)a * X_;
  float vals[4];
  float vmax = -INFINITY;
  #pragma unroll
  for (int i = 0; i < 4; ++i) {
    vals[i] = row[tid + i * 256];
    vmax = fmaxf(vmax, vals[i]);
  }
  vmax = block_max(vmax, red);
  float s = 0.f;
  #pragma unroll
  for (int i = 0; i < 4; ++i) {
    vals[i] = expf(vals[i] - vmax);
    s += vals[i];
  }
  s = block_sum(s, red);
  const float inv = 1.0f / s;
  #pragma unroll
  for (int i = 0; i < 4; ++i)
    score2[(size_t)a * X_ + tid + i * 256] = vals[i] * inv;
}

// ---------- Kernel B2: Wy[b,x] = Z[16+b,x] * score2[actions[b],x] + bias[x]
__global__ __launch_bounds__(256) void make_wy(const float* __restrict__ Z,
                                               const float* __restrict__ score2,
                                               const int* __restrict__ actions,
                                               const float* __restrict__ bias,
                                               float* __restrict__ Wy) {
  const int b = blockIdx.x;
  const int tid = threadIdx.x;
  const int act = actions[b];
  const float* yw = Z + (size_t)(A_ + b) * X_;
  const float* sc = score2 + (size_t)act * X_;
  #pragma unroll
  for (int i = 0; i < 4; ++i) {
    const int t = tid + i * 256;
    Wy[(size_t)b * X_ + t] = fmaf(yw[t], sc[t], bias[t]);
  }
}

// ---------- Kernel C (bandwidth dominant): dot[b,l] = sum_x x[b,l,x]*Wy[b,x]
// One block = 8 waves = 8 consecutive l of one batch; Wy[b] staged in LDS.
__global__ __launch_bounds__(256) void xwy_dot(const float* __restrict__ x,
                                               const float* __restrict__ Wy,
                                               const unsigned char* __restrict__ mask,
                                               float* __restrict__ dot) {
  __shared__ float wy_s[X_];
  const int b = blockIdx.y;
  const int tid = threadIdx.x;
  ((float4*)wy_s)[tid] = ((const float4*)(Wy + (size_t)b * X_))[tid];
  __syncthreads();

  const int lane = tid & 31;
  const int wv = tid >> 5;
  const int l = blockIdx.x * 8 + wv;
  const float4* __restrict__ xr =
      (const float4*)(x + ((size_t)b * L_ + l) * X_);
  const float4* wys4 = (const float4*)wy_s;

  float4 acc = make_float4(0.f, 0.f, 0.f, 0.f);
  #pragma unroll
  for (int i = 0; i < 8; ++i) {
    const float4 xv = xr[i * 32 + lane];   // global_load_b128, fully coalesced
    const float4 wv4 = wys4[i * 32 + lane];// ds_load_b128
    acc.x = fmaf(xv.x, wv4.x, acc.x);
    acc.y = fmaf(xv.y, wv4.y, acc.y);
    acc.z = fmaf(xv.z, wv4.z, acc.z);
    acc.w = fmaf(xv.w, wv4.w, acc.w);
  }
  float s = (acc.x + acc.y) + (acc.z + acc.w);
  #pragma unroll
  for (int off = 16; off; off >>= 1) s += __shfl_xor(s, off, 32);
  if (lane == 0) {
    if (mask[(size_t)b * L_ + l]) s = -INFINITY;
    dot[(size_t)b * L_ + l] = s;
  }
}

// ---------- Kernel D: out[b,:] = log_softmax(dot[b,:]) over L, one block per b
__global__ __launch_bounds__(256) void log_softmax_rows(const float* __restrict__ dot,
                                                        float* __restrict__ out) {
  __shared__ float red[8];
  const int b = blockIdx.x;
  const int tid = threadIdx.x;
  const float* row = dot + (size_t)b * L_;
  float vals[4];
  float vmax = -INFINITY;
  #pragma unroll
  for (int i = 0; i < 4; ++i) {
    vals[i] = row[tid + i * 256];
    vmax = fmaxf(vmax, vals[i]);
  }
  vmax = block_max(vmax, red);
  float s = 0.f;
  #pragma unroll
  for (int i = 0; i < 4; ++i) s += expf(vals[i] - vmax);
  s = block_sum(s, red);
  const float lse = logf(s);
  #pragma unroll
  for (int i = 0; i < 4; ++i)
    out[(size_t)b * L_ + tid + i * 256] = (vals[i] - vmax) - lse;
}

extern "C" void kernel_launch(void* const* d_in, const int* in_sizes, int n_in,
                              void* d_out, int out_size, void* d_ws, size_t ws_size,
                              hipStream_t stream) {
  const float* x               = (const float*)d_in[0];         // [B,L,X]
  const float* y               = (const float*)d_in[1];         // [B,Y]
  const unsigned char* x_mask  = (const unsigned char*)d_in[2]; // [B,L] bool
  const int* actions           = (const int*)d_in[3];           // [B]
  const float* weight          = (const float*)d_in[4];         // [Y,X]
  const float* bias            = (const float*)d_in[5];         // [X]
  const float* wa_h            = (const float*)d_in[6];         // [A,Y,1]
  float* out                   = (float*)d_out;                 // [B,L]

  // workspace layout (floats): Z[48*1024] | score2[16*1024] | Wy[32*1024] | dot[32*1024]
  float* Z      = (float*)d_ws;
  float* score2 = Z + (size_t)(A_ + B_) * X_;
  float* Wy     = score2 + (size_t)A_ * X_;
  float* dot    = Wy + (size_t)B_ * X_;

  gemm48_wmma<<<dim3(X_ / 16, 3), 32, 0, stream>>>(wa_h, y, weight, Z);
  softmax_rows<<<A_, 256, 0, stream>>>(Z, score2);
  make_wy<<<B_, 256, 0, stream>>>(Z, score2, actions, bias, Wy);
  xwy_dot<<<dim3(L_ / 8, B_), 256, 0, stream>>>(x, Wy, x_mask, dot);
  log_softmax_rows<<<B_, 256, 0, stream>>>(dot, out);
}